// CascadingSinkCache_26980984553670
// MI455X (gfx1250) — compile-verified
//
#include <hip/hip_runtime.h>
#include <hip/hip_bf16.h>
#include <stdint.h>

// Problem constants (mirror reference config)
#define Hh     32      // heads
#define Dd     128     // head dim
#define Ss     8192    // max_seq_len (cache slots)
#define Ww     512     // window length per cascade
#define Cc     16      // number of cascades
#define NSink  64      // sink tokens
#define Tt     1024    // streamed tokens

typedef float v4f __attribute__((ext_vector_type(4)));

// float4 counts / offsets within d_out (order: kc, vc, sc, sk, sv)
#define KC4   8388608LL                 // 32*8192*128/4
#define SC4_OFF   (2*KC4)               // 16777216
#define MAIN4 (2*KC4 + 2048LL)          // kc+vc+sc = 16779264 float4
#define SK4_OFF   16779264LL
#define SK4_CNT   65536LL               // 32*64*128/4
#define TOTAL_SINK4 (2*SK4_CNT)         // sk+sv = 131072 float4

// ---------------------------------------------------------------------------
// K1: sequential replay of the (data-independent) cache state machine.
// Produces slot_out[s] = token id held by cache slot s at the end, or -1.
// Scores are provably always zero, so b2's score gate always passes.
// ---------------------------------------------------------------------------
__global__ void sim_slots_kernel(int* __restrict__ slot_out) {
    __shared__ int slot[Ss];
    const int tid = threadIdx.x;
    for (int i = tid; i < Ss; i += blockDim.x) slot[i] = -1;
    __syncthreads();

    if (tid == 0) {
        int stored[Cc];
        int start[Cc];
        #pragma unroll
        for (int i = 0; i < Cc; ++i) { stored[i] = 0; start[i] = 0; }

        // tokens 0..NSink-1 go to sinks (handled analytically in K3).
        for (int t = NSink; t < Tt; ++t) {
            const int e = t - NSink;     // seen - 1 - num_sinks
            int  pend   = t;
            bool active = true;
            #pragma unroll
            for (int i = 0; i < Cc; ++i) {
                if (active) {
                    const bool dc  = ((e & ((1 << i) - 1)) == 0);
                    const int  st  = start[i];
                    const int  tok = stored[i];
                    const bool a1  = dc  && (tok <  Ww);
                    const bool a2  = dc  && (tok >= Ww);
                    const bool b1  = (!dc) && (tok == 0);
                    // b2 = !dc && tok>0 : always writes (scores all zero)
                    const int l = Ww * i;
                    int idx;
                    if (a1)            idx = l + st + tok;
                    else if (a2 || b1) idx = l + st;
                    else {
                        int r = (st - 1) % tok;
                        if (r < 0) r += tok;
                        idx = l + r;
                    }
                    const int ev = slot[idx];
                    slot[idx] = pend;
                    if (a1 || b1) stored[i] = tok + 1;
                    if (a2) { start[i] = (st + 1) & (Ww - 1); pend = ev; }
                    active = a2;       // eviction chain continues only on a2
                }
            }
        }
    }
    __syncthreads();
    for (int i = tid; i < Ss; i += blockDim.x) slot_out[i] = slot[i];
}

// ---------------------------------------------------------------------------
// K2: bulk fill of kc | vc | sc. One float4 per thread, exact coverage.
// Occupied slots gather the token's row from keys/values; everything else
// (empty slots + score_cache) is zero. All stores non-temporal: the 270 MB
// output is write-once streaming traffic, don't pollute L2.
// ---------------------------------------------------------------------------
__global__ void fill_main_kernel(const float* __restrict__ keys,
                                 const float* __restrict__ values,
                                 const int*   __restrict__ slot,
                                 float* __restrict__ out) {
    const long long g = (long long)blockIdx.x * blockDim.x + threadIdx.x;
    if (g >= MAIN4) return;
    v4f* out4 = (v4f*)out;
    v4f  val  = {0.0f, 0.0f, 0.0f, 0.0f};

    if (g < 2 * KC4) {
        const long long q   = (g < KC4) ? g : g - KC4;
        const float*    src = (g < KC4) ? keys : values;
        const int d4 = (int)(q & 31);
        const int s  = (int)((q >> 5) & (Ss - 1));
        const int h  = (int)(q >> 18);
        const int tok = slot[s];             // uniform per 32-lane wave
        if (tok >= 0) {
            const v4f* src4 = (const v4f*)src;
            val = src4[((long long)tok * Hh + h) * 32 + d4];
        }
    }
    __builtin_nontemporal_store(val, &out4[g]);
}

// ---------------------------------------------------------------------------
// K3: sink_keys | sink_values copy via the CDNA5 async DMA path:
//   global_load_async_to_lds_b128  ->  s_wait_asynccnt  ->
//   global_store_async_from_lds_b128 -> s_wait_asynccnt
// Sink slot i holds token i (T >= NSink). Exact coverage: 512*256 threads.
// ---------------------------------------------------------------------------
__global__ void sink_copy_async_kernel(const float* __restrict__ keys,
                                       const float* __restrict__ values,
                                       float* __restrict__ out) {
    __shared__ v4f stage[256];
    const int tid = threadIdx.x;
    const long long g = (long long)blockIdx.x * blockDim.x + tid; // [0, 131072)

    const bool       is_k = (g < SK4_CNT);
    const long long  q    = is_k ? g : g - SK4_CNT;
    const int d4   = (int)(q & 31);
    const int sidx = (int)((q >> 5) & (NSink - 1));
    const int h    = (int)(q >> 11);

    const v4f* src4 = (const v4f*)(is_k ? keys : values);
    const unsigned long long src_addr =
        (unsigned long long)&src4[((long long)sidx * Hh + h) * 32 + d4];
    const unsigned long long dst_addr =
        (unsigned long long)(((v4f*)out) + SK4_OFF + g);
    const unsigned lds_off = (unsigned)(unsigned long long)&stage[tid];

    // Global -> LDS (128b per lane), bypassing VGPR data path
    asm volatile("global_load_async_to_lds_b128 %0, %1, off"
                 :
                 : "v"(lds_off), "v"(src_addr)
                 : "memory");
    asm volatile("s_wait_asynccnt 0" ::: "memory");
    // LDS -> Global (128b per lane)
    asm volatile("global_store_async_from_lds_b128 %0, %1, off"
                 :
                 : "v"(dst_addr), "v"(lds_off)
                 : "memory");
    asm volatile("s_wait_asynccnt 0" ::: "memory");
}

// ---------------------------------------------------------------------------
extern "C" void kernel_launch(void* const* d_in, const int* in_sizes, int n_in,
                              void* d_out, int out_size, void* d_ws, size_t ws_size,
                              hipStream_t stream) {
    const float* keys   = (const float*)d_in[0];   // [T,B,H,1,D] f32
    const float* values = (const float*)d_in[1];   // [T,B,H,1,D] f32
    float*       out    = (float*)d_out;           // kc|vc|sc|sk|sv
    int*         slot   = (int*)d_ws;              // 8192 ints scratch

    sim_slots_kernel<<<1, 256, 0, stream>>>(slot);

    const int main_blocks = (int)((MAIN4 + 255) / 256);   // 65544
    fill_main_kernel<<<main_blocks, 256, 0, stream>>>(keys, values, slot, out);

    const int sink_blocks = (int)(TOTAL_SINK4 / 256);     // 512
    sink_copy_async_kernel<<<sink_blocks, 256, 0, stream>>>(keys, values, out);
}